// PoolingLayer_9320079032569
// MI455X (gfx1250) — compile-verified
//
#include <hip/hip_runtime.h>
#include <math.h>
#include <stdint.h>

#define D_MODEL 1024
#define SEQ 8192
#define BATCH 8
#define NH 16
#define EPS 1e-5f

typedef float v2f __attribute__((ext_vector_type(2)));
typedef float v8f __attribute__((ext_vector_type(8)));

// ---- workspace layout (float offsets) ----
#define OFF_QN     0
#define OFF_CONSTS (OFF_QN + 1024)
#define OFF_CK     (OFF_CONSTS + 16)              // float4[1024]: {gk*g, gk*b, gk, bk}
#define OFF_CV     (OFF_CK + 4096)                // float4[1024]: {gv*g, gv*b, gv, nv_b}
#define OFF_Q      (OFF_CV + 4096)                // q[1024]
#define OFF_QK     (OFF_Q + 1024)                 // qk[e][h] 16384
#define OFF_QB     (OFF_QK + 16384)               // qbias[16]
#define OFF_STATS  (OFF_QB + 16)                  // float4 per (b,s): m1,rs1,m2,rs2
#define OFF_SCORES (OFF_STATS + BATCH*SEQ*4)      // [B][S][H]
#define OFF_ATTN   (OFF_SCORES + BATCH*SEQ*NH)    // [B][H][S]
#define OFF_SUMA   (OFF_ATTN + BATCH*NH*SEQ)      // [B][H]
#define OFF_POOLED (OFF_SUMA + 128)               // [B][H][D]
#define OFF_OUT0   (OFF_POOLED + BATCH*NH*D_MODEL)// [B][D]

__device__ __forceinline__ float wsum(float v) {
#pragma unroll
  for (int m = 16; m >= 1; m >>= 1) v += __shfl_xor(v, m, 32);
  return v;
}

// ---------- setup: query LN, coeff packs, weight-sum constants ----------
__global__ void k_setup1(const float* __restrict__ query,
                         const float* __restrict__ norm_g, const float* __restrict__ norm_b,
                         const float* __restrict__ nq_g, const float* __restrict__ nq_b,
                         const float* __restrict__ nk_g, const float* __restrict__ nk_b,
                         const float* __restrict__ nv_g, const float* __restrict__ nv_b,
                         float* __restrict__ ws) {
  __shared__ float red[256];
  int tid = threadIdx.x;
  for (int e = tid; e < 1024; e += 256) {
    float g = norm_g[e], bb = norm_b[e];
    reinterpret_cast<float4*>(ws + OFF_CK)[e] = make_float4(nk_g[e]*g, nk_g[e]*bb, nk_g[e], nk_b[e]);
    reinterpret_cast<float4*>(ws + OFF_CV)[e] = make_float4(nv_g[e]*g, nv_g[e]*bb, nv_g[e], nv_b[e]);
  }
  float s1=0,s2=0,cg=0,cg2=0,cb=0,cb2=0,cgb=0;
  for (int e = tid; e < 1024; e += 256) {
    float q = query[e]; s1 += q; s2 += q*q;
    float g = norm_g[e], bb = norm_b[e];
    cg += g; cg2 += g*g; cb += bb; cb2 += bb*bb; cgb += g*bb;
  }
  float vals[7] = {s1,s2,cg,cg2,cb,cb2,cgb};
  for (int k = 0; k < 7; ++k) {
    red[tid] = vals[k]; __syncthreads();
    for (int off = 128; off >= 1; off >>= 1) {
      if (tid < off) red[tid] += red[tid+off];
      __syncthreads();
    }
    vals[k] = red[0]; __syncthreads();
  }
  float m = vals[0] * (1.f/1024.f);
  float rs = rsqrtf(vals[1]*(1.f/1024.f) - m*m + EPS);
  for (int e = tid; e < 1024; e += 256)
    ws[OFF_QN + e] = (query[e]-m)*rs*nq_g[e] + nq_b[e];
  if (tid < 5) ws[OFF_CONSTS + tid] = vals[2+tid];   // Cg,Cg2,Cb,Cb2,Cgb
}

// ---------- q = qn @ Wq^T + bq  (wave per output row) ----------
__global__ void k_qproj(const float* __restrict__ Wq, const float* __restrict__ bq,
                        float* __restrict__ ws) {
  int wave = threadIdx.x >> 5, lane = threadIdx.x & 31;
  int row = blockIdx.x * 8 + wave;
  const float* qn = ws + OFF_QN;
  const float* wr = Wq + (size_t)row * 1024;
  float s = 0;
  for (int e = lane; e < 1024; e += 32) s += wr[e]*qn[e];
  s = wsum(s);
  if (lane == 0) ws[OFF_Q + row] = s + bq[row];
}

// ---------- qk[e][h] = scale * sum_d q[h,d]*Wk[h*64+d][e];  qbias[h] ----------
__global__ void k_qk(const float* __restrict__ Wk, const float* __restrict__ bk,
                     float* __restrict__ ws) {
  int p = blockIdx.x * 256 + threadIdx.x;
  int h = p >> 10, e = p & 1023;
  const float* q = ws + OFF_Q + h*64;
  float s = 0;
  for (int d = 0; d < 64; ++d) s += q[d] * Wk[(size_t)(h*64+d)*1024 + e];
  const float scale = 0.125f;   // 1/sqrt(64)
  ws[OFF_QK + e*16 + h] = s * scale;
  if (e == 0) {
    float sb = 0;
    for (int d = 0; d < 64; ++d) sb += q[d] * bk[h*64+d];
    ws[OFF_QB + h] = sb * scale;
  }
}

// ---------- pass 1: per-row LN stats + WMMA scores [16 s-rows x 16 heads] ----------
__global__ void k_scores(const float* __restrict__ x,
                         const float* __restrict__ norm_g, const float* __restrict__ norm_b,
                         float* __restrict__ ws) {
  int wave = threadIdx.x >> 5, lane = threadIdx.x & 31;
  int tile = blockIdx.x * 4 + wave;
  int b = tile >> 9;
  int sb = (tile & 511) << 4;
  int r = lane & 15, hf = lane >> 4;
  const float Cg = ws[OFF_CONSTS+0], Cg2 = ws[OFF_CONSTS+1],
              Cb = ws[OFF_CONSTS+2], Cb2 = ws[OFF_CONSTS+3], Cgb = ws[OFF_CONSTS+4];
  float mm1=0, rr1=0, mm2=0, rr2=0;
  for (int rr = 0; rr < 16; ++rr) {
    const float* xr = x + (size_t)(b*SEQ + sb + rr) * 1024;
    float S1=0,S2=0,Sg=0,Sg2x=0,Sg2x2=0,Sgbx=0;
    for (int e = lane; e < 1024; e += 32) {
      float xv = xr[e], g = norm_g[e], bb = norm_b[e];
      float gx = g*xv;
      S1 += xv; S2 += xv*xv; Sg += gx; Sg2x += g*gx; Sg2x2 += gx*gx; Sgbx += bb*gx;
    }
    S1 = wsum(S1); S2 = wsum(S2); Sg = wsum(Sg);
    Sg2x = wsum(Sg2x); Sg2x2 = wsum(Sg2x2); Sgbx = wsum(Sgbx);
    float m1 = S1 * (1.f/1024.f);
    float rs1 = rsqrtf(S2*(1.f/1024.f) - m1*m1 + EPS);
    float m2 = (rs1*(Sg - m1*Cg) + Cb) * (1.f/1024.f);
    float ex2 = rs1*rs1*(Sg2x2 - 2.f*m1*Sg2x + m1*m1*Cg2) + 2.f*rs1*(Sgbx - m1*Cgb) + Cb2;
    float rs2 = rsqrtf(ex2*(1.f/1024.f) - m2*m2 + EPS);
    if (lane == rr)
      reinterpret_cast<float4*>(ws + OFF_STATS)[b*SEQ + sb + rr] = make_float4(m1, rs1, m2, rs2);
    if (r == rr) { mm1=m1; rr1=rs1; mm2=m2; rr2=rs2; }
  }
  // phase 2: scores tile via V_WMMA_F32_16X16X4_F32, K-loop over D
  const float* xrow = x + (size_t)(b*SEQ + sb + r) * 1024;
  const float4* ck4 = reinterpret_cast<const float4*>(ws + OFF_CK);
  const float* qk = ws + OFF_QK;
  v8f c = {0.f,0.f,0.f,0.f,0.f,0.f,0.f,0.f};
  for (int e = 0; e < 1024; e += 4) {
    int eA = e + 2*hf;
    float2 xv = *reinterpret_cast<const float2*>(xrow + eA);
    float4 c0 = ck4[eA], c1 = ck4[eA+1];
    float kn0 = rr2*((xv.x - mm1)*rr1*c0.x + c0.y - mm2*c0.z) + c0.w;
    float kn1 = rr2*((xv.y - mm1)*rr1*c1.x + c1.y - mm2*c1.z) + c1.w;
    v2f a = {kn0, kn1};
    v2f bm = {qk[eA*16 + r], qk[(eA+1)*16 + r]};
    c = __builtin_amdgcn_wmma_f32_16x16x4_f32(false, a, false, bm, (short)0, c, false, false);
  }
  float qb = ws[OFF_QB + r];
  float* sc = ws + OFF_SCORES;
#pragma unroll
  for (int j = 0; j < 8; ++j) {
    int s = sb + j + 8*hf;
    sc[(size_t)(b*SEQ + s)*16 + r] = c[j] + qb;
  }
}

// ---------- softmax per (b,h) with clamp; writes attn [B][H][S] and sum(attn) ----------
__global__ void k_softmax(float* __restrict__ ws) {
  __shared__ float red[256];
  int tid = threadIdx.x;
  int b = blockIdx.x >> 4, h = blockIdx.x & 15;
  const float* sc = ws + OFF_SCORES + (size_t)b*SEQ*16 + h;
  float mx = -INFINITY;
  for (int s = tid; s < SEQ; s += 256) mx = fmaxf(mx, sc[(size_t)s*16]);
  red[tid] = mx; __syncthreads();
  for (int off=128; off>=1; off>>=1){ if(tid<off) red[tid]=fmaxf(red[tid],red[tid+off]); __syncthreads(); }
  mx = red[0]; __syncthreads();
  float sum = 0;
  for (int s = tid; s < SEQ; s += 256) sum += __expf(sc[(size_t)s*16] - mx);
  red[tid] = sum; __syncthreads();
  for (int off=128; off>=1; off>>=1){ if(tid<off) red[tid]+=red[tid+off]; __syncthreads(); }
  float inv = 1.f / (red[0] + 1e-9f); __syncthreads();
  float* at = ws + OFF_ATTN + (size_t)(b*16 + h)*SEQ;
  float sa = 0;
  for (int s = tid; s < SEQ; s += 256) {
    float a = __expf(sc[(size_t)s*16] - mx) * inv;
    a = fminf(fmaxf(a, 1e-9f), 1.f);
    at[s] = a; sa += a;
  }
  red[tid] = sa; __syncthreads();
  for (int off=128; off>=1; off>>=1){ if(tid<off) red[tid]+=red[tid+off]; __syncthreads(); }
  if (tid == 0) ws[OFF_SUMA + b*16 + h] = red[0];
}

// ---------- pass 2: pooled[b,h,e] = sum_s attn * vn  via WMMA (M=h, N=e-tile, K=s) ----------
// x tiles staged through LDS with GLOBAL_LOAD_ASYNC_TO_LDS_B128, per-wave double buffer,
// synchronized with s_wait_asynccnt (async loads complete in order per wave).
#define POOL_STAGE_ROWS 32
#define POOL_STAGES     (SEQ / 4 / POOL_STAGE_ROWS)   // 2048 rows per wave / 32 = 64
__global__ void k_pool(const float* __restrict__ x, float* __restrict__ ws) {
  __shared__ __align__(16) float xbuf[4][2][POOL_STAGE_ROWS * 16]; // 4 waves x 2 bufs x 2KB
  __shared__ float tile[256];
  int wave = threadIdx.x >> 5, lane = threadIdx.x & 31;
  int b = blockIdx.x >> 6, e0 = (blockIdx.x & 63) << 4;
  int r = lane & 15, hf = lane >> 4;
  int col = e0 + r;
  float4 cv = reinterpret_cast<const float4*>(ws + OFF_CV)[col];
  const float* at = ws + OFF_ATTN + (size_t)(b*16 + r)*SEQ;
  const float4* st4 = reinterpret_cast<const float4*>(ws + OFF_STATS) + (size_t)b*SEQ;
  const float* xb = x + (size_t)b*SEQ*1024;
  const uint64_t gbase = (uint64_t)(uintptr_t)xb + (uint64_t)((unsigned)e0 * 4u);
  const unsigned ldsbase = (unsigned)(uintptr_t)&xbuf[wave][0][0];
  const int sw0 = wave * (SEQ/4);              // this wave's contiguous row span

  // issue one 32-row stage: each lane DMAs its own row's 64B as 4x b128
  auto issue = [&](int st, int bufidx) {
    unsigned row = (unsigned)(sw0 + st*POOL_STAGE_ROWS + lane);
    uint64_t ga = gbase + (uint64_t)(row * 4096u);
    unsigned la = ldsbase + (unsigned)bufidx*(POOL_STAGE_ROWS*16*4) + (unsigned)lane*64u;
#pragma unroll
    for (int p = 0; p < 4; ++p) {
      asm volatile("global_load_async_to_lds_b128 %0, %1, off"
                   :: "v"(la + (unsigned)(p*16)), "v"(ga + (uint64_t)(p*16)) : "memory");
    }
  };

  v8f acc = {0.f,0.f,0.f,0.f,0.f,0.f,0.f,0.f};
  issue(0, 0);
  for (int st = 0; st < POOL_STAGES; ++st) {
    int cur = st & 1;
    if (st + 1 < POOL_STAGES) {
      issue(st + 1, cur ^ 1);
      asm volatile("s_wait_asynccnt 0x4" ::: "memory");  // stage st's 4 ops complete
    } else {
      asm volatile("s_wait_asynccnt 0x0" ::: "memory");
    }
    const float* buf = &xbuf[wave][cur][0];
#pragma unroll
    for (int k = 0; k < POOL_STAGE_ROWS/4; ++k) {
      int lrow = k*4 + 2*hf;                          // row within stage for this lane
      int sA = sw0 + st*POOL_STAGE_ROWS + lrow;       // absolute row
      float2 av = *reinterpret_cast<const float2*>(at + sA);
      float4 t0 = st4[sA], t1 = st4[sA+1];
      float x0 = buf[lrow*16 + r];
      float x1 = buf[(lrow+1)*16 + r];
      float vn0 = t0.w*((x0 - t0.x)*t0.y*cv.x + cv.y - t0.z*cv.z) + cv.w;
      float vn1 = t1.w*((x1 - t1.x)*t1.y*cv.x + cv.y - t1.z*cv.z) + cv.w;
      v2f a = {av.x, av.y};
      v2f bm = {vn0, vn1};
      acc = __builtin_amdgcn_wmma_f32_16x16x4_f32(false, a, false, bm, (short)0, acc, false, false);
    }
  }
  tile[threadIdx.x] = 0.f; tile[threadIdx.x + 128] = 0.f;
  __syncthreads();
#pragma unroll
  for (int j = 0; j < 8; ++j)
    atomicAdd(&tile[(j + 8*hf)*16 + r], acc[j]);
  __syncthreads();
  float* pooled = ws + OFF_POOLED;
  for (int k = threadIdx.x; k < 256; k += 128) {
    int h = k >> 4, cc = k & 15;
    pooled[(size_t)(b*16 + h)*1024 + e0 + cc] = tile[k];
  }
}

// ---------- out0[b,j] = Wv[j,:]·pooled[b,h(j),:] + sum(attn)*bv[j] ----------
__global__ void k_attnout(const float* __restrict__ Wv, const float* __restrict__ bv,
                          float* __restrict__ ws) {
  int wave = threadIdx.x >> 5, lane = threadIdx.x & 31;
  int row = blockIdx.x * 8 + wave;
  int b = row >> 10, j = row & 1023;
  int h = j >> 6;
  const float* wr = Wv + (size_t)j * 1024;
  const float* pl = ws + OFF_POOLED + (size_t)(b*16 + h)*1024;
  float s = 0;
  for (int e = lane; e < 1024; e += 32) s += wr[e] * pl[e];
  s = wsum(s);
  if (lane == 0) ws[OFF_OUT0 + row] = s + ws[OFF_SUMA + b*16 + h] * bv[j];
}

// ---------- final: y = Wo·out0 + bo, then LayerNorm(no_g, no_b) ----------
__global__ void k_outproj(const float* __restrict__ Wo, const float* __restrict__ bo,
                          const float* __restrict__ no_g, const float* __restrict__ no_b,
                          const float* __restrict__ ws, float* __restrict__ out) {
  __shared__ float y[1024];
  __shared__ float red[256];
  int tid = threadIdx.x, wave = tid >> 5, lane = tid & 31;
  int b = blockIdx.x;
  const float* o0 = ws + OFF_OUT0 + b*1024;
  for (int it = 0; it < 128; ++it) {
    int j = it*8 + wave;
    const float* wr = Wo + (size_t)j*1024;
    float s = 0;
    for (int i = lane; i < 1024; i += 32) s += wr[i]*o0[i];
    s = wsum(s);
    if (lane == 0) y[j] = s + bo[j];
  }
  __syncthreads();
  float s1=0, s2=0;
  for (int j = tid; j < 1024; j += 256) { float v = y[j]; s1 += v; s2 += v*v; }
  red[tid]=s1; __syncthreads();
  for (int off=128; off>=1; off>>=1){ if(tid<off) red[tid]+=red[tid+off]; __syncthreads(); }
  s1 = red[0]; __syncthreads();
  red[tid]=s2; __syncthreads();
  for (int off=128; off>=1; off>>=1){ if(tid<off) red[tid]+=red[tid+off]; __syncthreads(); }
  s2 = red[0]; __syncthreads();
  float m = s1*(1.f/1024.f);
  float rs = rsqrtf(s2*(1.f/1024.f) - m*m + EPS);
  for (int j = tid; j < 1024; j += 256)
    out[b*1024 + j] = (y[j]-m)*rs*no_g[j] + no_b[j];
}

extern "C" void kernel_launch(void* const* d_in, const int* in_sizes, int n_in,
                              void* d_out, int out_size, void* d_ws, size_t ws_size,
                              hipStream_t stream) {
  const float* x      = (const float*)d_in[0];
  const float* query  = (const float*)d_in[1];
  const float* norm_g = (const float*)d_in[2];
  const float* norm_b = (const float*)d_in[3];
  const float* nq_g   = (const float*)d_in[4];
  const float* nq_b   = (const float*)d_in[5];
  const float* nk_g   = (const float*)d_in[6];
  const float* nk_b   = (const float*)d_in[7];
  const float* nv_g   = (const float*)d_in[8];
  const float* nv_b   = (const float*)d_in[9];
  const float* no_g   = (const float*)d_in[10];
  const float* no_b   = (const float*)d_in[11];
  const float* Wq = (const float*)d_in[12]; const float* bq = (const float*)d_in[13];
  const float* Wk = (const float*)d_in[14]; const float* bk = (const float*)d_in[15];
  const float* Wv = (const float*)d_in[16]; const float* bv = (const float*)d_in[17];
  const float* Wo = (const float*)d_in[18]; const float* bo = (const float*)d_in[19];
  float* ws  = (float*)d_ws;
  float* out = (float*)d_out;

  k_setup1 <<<1,   256, 0, stream>>>(query, norm_g, norm_b, nq_g, nq_b, nk_g, nk_b, nv_g, nv_b, ws);
  k_qproj  <<<128, 256, 0, stream>>>(Wq, bq, ws);
  k_qk     <<<64,  256, 0, stream>>>(Wk, bk, ws);
  k_scores <<<BATCH*SEQ/16/4, 128, 0, stream>>>(x, norm_g, norm_b, ws);
  k_softmax<<<BATCH*NH, 256, 0, stream>>>(ws);
  k_pool   <<<BATCH*64, 128, 0, stream>>>(x, ws);
  k_attnout<<<BATCH*1024/8, 256, 0, stream>>>(Wv, bv, ws);
  k_outproj<<<BATCH, 256, 0, stream>>>(Wo, bo, no_g, no_b, ws, out);
}